// ICSBP_13958643712718
// MI455X (gfx1250) — compile-verified
//
#include <hip/hip_runtime.h>

#define FEATN 128
#define LATN  64
#define IMGN  64
#define PIXN  4096   // 64*64
#define BSN   64
#define NSTEPS 6

typedef __attribute__((ext_vector_type(2))) float v2f;
typedef __attribute__((ext_vector_type(8))) float v8f;

// ---------------------------------------------------------------------------
// Kernel 1: c_f = gate*(W @ x + b) + uv   (and delta = pre-uv channels 62,63)
// One wave computes a 16(M=LAT chans) x 16(N=pixels) f32 tile via
// v_wmma_f32_16x16x4_f32, K=128 in 32 steps. Weights staged in LDS with
// pitch 129 floats so the 16 A-lanes hit 16 distinct banks.
// ---------------------------------------------------------------------------
__global__ __launch_bounds__(256) void ICSBP_gemm_cf_kernel(
    const float* __restrict__ x,       // (BS, FEAT, 64, 64)
    const float* __restrict__ conv_w,  // (LAT, FEAT)
    const float* __restrict__ conv_b,  // (LAT,)
    const float* __restrict__ gate_p,  // scalar
    float* __restrict__ c_f,           // (BS, LAT, 64, 64)
    float* __restrict__ delta)         // (BS, 2, 64, 64)
{
    __shared__ float lds_w[LATN * 129];

    const int tid = threadIdx.x;
    // cooperative stage of conv_w (64x128 f32 = 32KB) into padded LDS
    for (int i = tid; i < LATN * FEATN; i += 256) {
        int o = i >> 7, f = i & 127;
        lds_w[o * 129 + f] = conv_w[i];
    }
    __syncthreads();

    const int bid    = blockIdx.x;      // 8192 blocks: 64 batches * 128
    const int batch  = bid >> 7;
    const int pblk   = (bid & 127) * 32;  // 32 pixels per block
    const int wave   = tid >> 5;
    const int lane   = tid & 31;
    const int m_tile = wave & 3;          // which 16 channels
    const int pgrp   = wave >> 2;         // which 16-pixel half
    const int p_base = pblk + pgrp * 16;  // pixel base within batch
    const int n      = lane & 15;
    const int khalf  = (lane >> 4) ? 2 : 0;  // K sub-offset per ISA A/B layout

    const float* xb   = x + (size_t)batch * FEATN * PIXN + p_base + n;
    const float* wrow = &lds_w[(m_tile * 16 + n) * 129 + khalf];

    v8f acc = {};
    #pragma unroll 8
    for (int k0 = 0; k0 < FEATN; k0 += 4) {
        v2f a, b;
        // A (16x4 f32): lane M = lane%16; VGPR0 holds K=k0+khalf, VGPR1 K=k0+khalf+1
        a.x = wrow[k0];
        a.y = wrow[k0 + 1];
        // B (4x16 f32): row K striped across lanes within a VGPR, mirrored halves
        b.x = xb[(size_t)(k0 + khalf) * PIXN];
        b.y = xb[(size_t)(k0 + khalf + 1) * PIXN];
        acc = __builtin_amdgcn_wmma_f32_16x16x4_f32(
            /*neg_a=*/false, a, /*neg_b=*/false, b,
            /*c_mod=*/(short)0, acc, /*reuse_a=*/false, /*reuse_b=*/false);
    }

    const float gate = *gate_p;
    const int p = p_base + n;           // pixel index within batch
    const int h = p >> 6, w = p & 63;
    const float g1 = -1.0f + 2.0f * (float)h * (1.0f / 63.0f);
    const float g2 = -1.0f + 2.0f * (float)w * (1.0f / 63.0f);
    const int mhi = (lane >> 4) ? 8 : 0;

    #pragma unroll
    for (int r = 0; r < 8; r++) {
        const int o = m_tile * 16 + r + mhi;          // output channel
        const float val = gate * (acc[r] + conv_b[o]); // pre-uv
        float cf = val;
        if (o == LATN - 2) cf += g1;
        if (o == LATN - 1) cf += g2;
        c_f[((size_t)batch * LATN + o) * PIXN + p] = cf;
        if (o == LATN - 2) delta[((size_t)batch * 2 + 0) * PIXN + p] = val;
        if (o == LATN - 1) delta[((size_t)batch * 2 + 1) * PIXN + p] = val;
    }
}

// ---------------------------------------------------------------------------
// Kernel 2: 6 sequential clustering steps, one workgroup per batch.
// 1024 threads (32 waves/WGP for max memory-level parallelism); each thread
// owns 4 consecutive pixels, so log_s lives in 4 registers per thread and
// the distance loop reads c_f with fully-coalesced b128 loads.
// ---------------------------------------------------------------------------
__global__ __launch_bounds__(1024) void ICSBP_cluster_kernel(
    const float* __restrict__ rand_pixel,   // (BS, 1, 64, 64)
    const float* __restrict__ log_sigma_p,  // scalar
    const float* __restrict__ c_f,          // (BS, LAT, 64, 64)
    float* __restrict__ log_m_k,            // (7, BS, 1, 64, 64)
    float* __restrict__ log_s_k,            // (7, BS, 1, 64, 64)
    float* __restrict__ seeds)              // (6, BS, LAT)
{
    __shared__ float seed[LATN];
    __shared__ float rv[1024];
    __shared__ int   ri[1024];

    const int b   = blockIdx.x;
    const int tid = threadIdx.x;
    const int p0  = tid * 4;                 // 4 consecutive pixels per thread
    const float sigma_inv = 1.0f / __expf(*log_sigma_p);
    const float* cfb = c_f + (size_t)b * LATN * PIXN;

    const float4 rp4 = *(const float4*)(rand_pixel + (size_t)b * PIXN + p0);
    const float rp[4] = {rp4.x, rp4.y, rp4.z, rp4.w};
    float ls[4] = {0.0f, 0.0f, 0.0f, 0.0f};

    // log_s_k[0] = zeros
    *(float4*)(log_s_k + (size_t)b * PIXN + p0) = make_float4(0.f, 0.f, 0.f, 0.f);

    for (int step = 0; step < NSTEPS; step++) {
        // ---- argmax over rand_pixel * exp(log_s) (first-index tie-break) ----
        float bv = -1e30f;
        int   bi = p0;
        #pragma unroll
        for (int j = 0; j < 4; j++) {
            float pv = rp[j] * __expf(ls[j]);
            if (pv > bv) { bv = pv; bi = p0 + j; }
        }
        rv[tid] = bv; ri[tid] = bi;
        __syncthreads();
        for (int s = 512; s > 0; s >>= 1) {
            if (tid < s) {
                float ov = rv[tid + s];
                int   oi = ri[tid + s];
                if (ov > rv[tid] || (ov == rv[tid] && oi < ri[tid])) {
                    rv[tid] = ov; ri[tid] = oi;
                }
            }
            __syncthreads();
        }
        const int idx = ri[0];

        // ---- gather seed (64 channels) ----
        if (tid < LATN) {
            float sv = cfb[(size_t)tid * PIXN + idx];
            seed[tid] = sv;
            seeds[((size_t)step * BSN + b) * LATN + tid] = sv;
        }
        __syncthreads();

        // ---- distance: b128 loads, 4 pixels per thread ----
        float d2[4] = {0.f, 0.f, 0.f, 0.f};
        #pragma unroll 8
        for (int o = 0; o < LATN; o++) {
            float4 c4 = *(const float4*)(cfb + (size_t)o * PIXN + p0);
            float sv = seed[o];
            float dx = c4.x - sv, dy = c4.y - sv, dz = c4.z - sv, dw = c4.w - sv;
            d2[0] = fmaf(dx, dx, d2[0]);
            d2[1] = fmaf(dy, dy, d2[1]);
            d2[2] = fmaf(dz, dz, d2[2]);
            d2[3] = fmaf(dw, dw, d2[3]);
        }

        float lm[4];
        #pragma unroll
        for (int j = 0; j < 4; j++) {
            float dd    = fminf(fmaxf(d2[j], 1e-10f), 1e10f);
            float alpha = __expf(-sqrtf(dd) * sigma_inv);
            alpha = fminf(fmaxf(alpha, 0.01f), 0.99f);
            lm[j] = ls[j] + __logf(alpha);
            ls[j] = ls[j] + __logf(1.0f - alpha);
        }
        *(float4*)(log_m_k + ((size_t)step * BSN + b) * PIXN + p0) =
            make_float4(lm[0], lm[1], lm[2], lm[3]);
        *(float4*)(log_s_k + ((size_t)(step + 1) * BSN + b) * PIXN + p0) =
            make_float4(ls[0], ls[1], ls[2], ls[3]);
        // seed[] reads above are ordered before the next step's seed write by
        // the argmax reduction's __syncthreads chain.
    }

    // log_m_k[6] = final log_s
    *(float4*)(log_m_k + ((size_t)NSTEPS * BSN + b) * PIXN + p0) =
        make_float4(ls[0], ls[1], ls[2], ls[3]);
}

extern "C" void kernel_launch(void* const* d_in, const int* in_sizes, int n_in,
                              void* d_out, int out_size, void* d_ws, size_t ws_size,
                              hipStream_t stream) {
    const float* x          = (const float*)d_in[0];
    const float* rand_pixel = (const float*)d_in[1];
    const float* conv_w     = (const float*)d_in[2];
    const float* conv_b     = (const float*)d_in[3];
    const float* gate       = (const float*)d_in[4];
    const float* log_sigma  = (const float*)d_in[5];
    // d_in[6] = num_cl_steps == 6 (compile-time constant here)

    float* out = (float*)d_out;
    const size_t SZ_MK = (size_t)(NSTEPS + 1) * BSN * PIXN;   // 1,835,008
    float* log_m_k = out;
    float* log_s_k = out + SZ_MK;
    float* c_f     = log_s_k + SZ_MK;                         // 16,777,216
    float* delta   = c_f + (size_t)BSN * LATN * PIXN;         // 524,288
    float* seeds   = delta + (size_t)BSN * 2 * PIXN;          // 24,576

    ICSBP_gemm_cf_kernel<<<8192, 256, 0, stream>>>(x, conv_w, conv_b, gate, c_f, delta);
    ICSBP_cluster_kernel<<<BSN, 1024, 0, stream>>>(rand_pixel, log_sigma, c_f,
                                                   log_m_k, log_s_k, seeds);
}